// DiagonalMultiplySum_21715354648728
// MI455X (gfx1250) — compile-verified
//
#include <hip/hip_runtime.h>

// out[b,o,s] = sum_i in[b,i,s] * diag[o,i,s]
// 4096 per-position GEMMs (M=64,N=256,K=256), fp32, V_WMMA_F32_16X16X4_F32.
// Wave owns one s and a 2x2 grid of 16x16 tiles; staging is double-buffered in
// registers so global-load latency hides behind the previous chunk's WMMAs.

typedef float v2f __attribute__((ext_vector_type(2)));
typedef float v4f __attribute__((ext_vector_type(4)));
typedef float v8f __attribute__((ext_vector_type(8)));

constexpr int OUT_C = 256;
constexpr int IN_C  = 256;
constexpr int SIZE  = 4096;

constexpr int SB = 32;          // s-values per workgroup (128B contiguous global rows)
constexpr int KC = 8;           // K staged per LDS chunk
constexpr int KS = 41;          // LDS k-stride (32 m + 9 pad): bank-conflict-free
constexpr int SS = KC * KS + 1; // 329 = LDS s-stride (odd mod 4 -> half-lanes +16 banks)
constexpr int MATF = SB * SS;   // 10528 floats per staged matrix (~41 KB)

// LDS chunk layout per matrix: idx = s*SS + k*KS + m   (m in 0..31)

__global__ __launch_bounds__(1024)
void DiagonalMultiplySum_kernel(const float* __restrict__ inp,
                                const float* __restrict__ diag,
                                float* __restrict__ out) {
    extern __shared__ float lds[];          // [2][MATF]: A then B
    const float* As = lds;
    const float* Bs = lds + MATF;

    const int tid  = threadIdx.x;
    const int lane = tid & 31;
    const int wave = tid >> 5;              // 0..31 == this wave's s within the block

    // grid: x = (2 m-groups x 8 n-groups), y = s-block (slow axis -> L2 slab reuse)
    const int mblk = blockIdx.x >> 3;       // 0..1
    const int nblk = blockIdx.x & 7;        // 0..7
    const int b0 = mblk * 32;
    const int o0 = nblk * 32;
    const int s0 = blockIdx.y * SB;

    // ---- cooperative loader role: 2 matrices x 256 rows x 2 half-rows = 1024 ----
    const int isB  = tid >> 9;              // uniform per wave: waves 0-15 -> A, 16-31 -> B
    const int row  = (tid >> 1) & 255;
    const int mi   = row >> 3;              // 0..31 (m for A, n for B)
    const int ki   = row & 7;               // 0..7
    const int soff = (tid & 1) * 16;        // which 64B half of the 128B s-row

    const float* gbase = isB
        ? diag + (size_t)(o0 + mi) * (IN_C * SIZE)
        : inp  + (size_t)(b0 + mi) * (IN_C * SIZE);
    const float* gp = gbase + (size_t)ki * SIZE + s0 + soff;
    float*       lp = lds + isB * MATF + ki * KS + mi;

    // ---- WMMA lane roles (16x16x4 f32) ----
    const int lh = lane >> 4;               // 0: K=0,1   1: K=2,3
    const int lm = lane & 15;               // M (A) / N (B) within tile
    const int sbase = wave * SS;

    v8f acc[2][2];
#pragma unroll
    for (int mt = 0; mt < 2; ++mt)
#pragma unroll
        for (int nt = 0; nt < 2; ++nt)
            acc[mt][nt] = (v8f){};

    // one chunk of staged data held in registers (4 x v4f = 64B half-row)
    auto gload = [&](v4f q[4], int kc) {
        const float* p = gp + (size_t)kc * SIZE;
        q[0] = *(const v4f*)(p + 0);
        q[1] = *(const v4f*)(p + 4);
        q[2] = *(const v4f*)(p + 8);
        q[3] = *(const v4f*)(p + 12);
    };
    auto lcommit = [&](const v4f q[4]) {
#pragma unroll
        for (int e = 0; e < 4; ++e) {
            lp[(soff + e +  0) * SS] = q[0][e];
            lp[(soff + e +  4) * SS] = q[1][e];
            lp[(soff + e +  8) * SS] = q[2][e];
            lp[(soff + e + 12) * SS] = q[3][e];
        }
    };
    auto compute = [&]() {
#pragma unroll
        for (int k = 0; k < KC; k += 4) {
            const int ke = k + lh * 2;
            v2f a[2], b[2];
#pragma unroll
            for (int mt = 0; mt < 2; ++mt) {
                a[mt].x = As[sbase + (ke    ) * KS + mt * 16 + lm];
                a[mt].y = As[sbase + (ke + 1) * KS + mt * 16 + lm];
            }
#pragma unroll
            for (int nt = 0; nt < 2; ++nt) {
                b[nt].x = Bs[sbase + (ke    ) * KS + nt * 16 + lm];
                b[nt].y = Bs[sbase + (ke + 1) * KS + nt * 16 + lm];
            }
#pragma unroll
            for (int mt = 0; mt < 2; ++mt)
#pragma unroll
                for (int nt = 0; nt < 2; ++nt)
                    acc[mt][nt] = __builtin_amdgcn_wmma_f32_16x16x4_f32(
                        false, a[mt], false, b[nt], (short)0, acc[mt][nt],
                        false, false);
        }
    };

    // ---- software-pipelined chunk loop (ping/pong register staging) ----
    v4f qa[4], qb[4];
    gload(qa, 0);                            // prologue: chunk 0 in flight

    for (int kc = 0; kc < IN_C; kc += 2 * KC) {
        __syncthreads();                     // LDS free (prev compute done)
        lcommit(qa);                         // waits only on qa's own loads
        gload(qb, kc + KC);                  // next chunk in flight during compute
        __syncthreads();
        compute();                           // chunk kc

        __syncthreads();
        lcommit(qb);
        if (kc + 2 * KC < IN_C)              // uniform scalar branch
            gload(qa, kc + 2 * KC);
        __syncthreads();
        compute();                           // chunk kc + KC
    }

    // ---- writeback: C/D layout vgpr j -> M = j + 8*lh, N = lm ----
    // The 32 waves jointly cover s0..s0+31 per (b,o); 4B stores merge in L2.
#pragma unroll
    for (int mt = 0; mt < 2; ++mt)
#pragma unroll
        for (int nt = 0; nt < 2; ++nt) {
            const int o = o0 + nt * 16 + lm;
#pragma unroll
            for (int j = 0; j < 8; ++j) {
                const int b = b0 + mt * 16 + j + 8 * lh;
                out[(size_t)b * (OUT_C * SIZE) + (size_t)o * SIZE + s0 + wave] =
                    acc[mt][nt][j];
            }
        }
}

extern "C" void kernel_launch(void* const* d_in, const int* in_sizes, int n_in,
                              void* d_out, int out_size, void* d_ws, size_t ws_size,
                              hipStream_t stream) {
    const float* inp  = (const float*)d_in[0];   // [64, 256, 4096]
    const float* diag = (const float*)d_in[1];   // [256, 256, 4096]
    float* out = (float*)d_out;                  // [64, 256, 4096]

    dim3 grid(2 * 8, SIZE / SB);                 // 16 tile-groups x 128 s-blocks
    dim3 block(1024);                            // 32 wave32
    size_t shmem = 2u * MATF * sizeof(float);    // ~82 KB dynamic LDS (320 KB WGP pool)

    hipLaunchKernelGGL(DiagonalMultiplySum_kernel, grid, block, shmem, stream,
                       inp, diag, out);
}